// CausalSelfAttention_30958124270147
// MI455X (gfx1250) — compile-verified
//
#include <hip/hip_runtime.h>

typedef float v2f __attribute__((ext_vector_type(2)));
typedef float v8f __attribute__((ext_vector_type(8)));

// ---------------------------------------------------------------------------
// Generic fp32 GEMM: C[M,N] = A[M,K] * B[K,N], row-major, using
// V_WMMA_F32_16X16X4_F32.  Block = 256 threads = 8 waves.
// Block tile 64(M) x 128(N); wave tile 16(M) x 64(N); K staged 16 at a time.
// B is stored pair-interleaved in LDS so each WMMA B-fragment is one
// aligned ds_load_b64 (the fragment needs {B[k][n], B[k+1][n]} per lane).
// ---------------------------------------------------------------------------
#define GM_TILE_M 64
#define GM_TILE_N 128
#define GM_KC     16

__global__ __launch_bounds__(256)
void gemm_f32_wmma(const float* __restrict__ A, const float* __restrict__ B,
                   float* __restrict__ C, int M, int N, int K) {
    __shared__ float As[GM_TILE_M][GM_KC + 4];          // row-major, padded
    __shared__ v2f   Bs[GM_KC / 2][GM_TILE_N + 4];      // pair-interleaved

    const int tid  = threadIdx.x;
    const int lane = tid & 31;
    const int wave = tid >> 5;
    const int wm   = wave >> 1;          // 0..3  (M sub-tile)
    const int wn   = wave & 1;           // 0..1  (N sub-tile)
    const int l16  = lane & 15;
    const int hi   = lane >> 4;
    const int m0   = blockIdx.y * GM_TILE_M;
    const int n0   = blockIdx.x * GM_TILE_N;

    v8f acc[4];
#pragma unroll
    for (int i = 0; i < 4; ++i) acc[i] = (v8f){};

    for (int k0 = 0; k0 < K; k0 += GM_KC) {
        // Stage A tile: 64x16 floats, one float4 per thread (row-major).
        {
            const int r = tid >> 2;              // 0..63
            const int c = (tid & 3) << 2;        // 0,4,8,12
            const float4 av = *(const float4*)(A + (size_t)(m0 + r) * K + k0 + c);
            *(float4*)(&As[r][c]) = av;
        }
        // Stage B tile: 16x128 floats, pair-interleaved across k.
        // Thread owns row-pair p = tid>>5 and 4 columns.
        {
            const int p = tid >> 5;              // 0..7  (k row pair)
            const int c = (tid & 31) << 2;       // 0..124 step 4
            const float* g0 = B + (size_t)(k0 + 2 * p) * N + n0 + c;
            const float4 lo = *(const float4*)(g0);
            const float4 hv = *(const float4*)(g0 + N);
            Bs[p][c + 0] = (v2f){lo.x, hv.x};
            Bs[p][c + 1] = (v2f){lo.y, hv.y};
            Bs[p][c + 2] = (v2f){lo.z, hv.z};
            Bs[p][c + 3] = (v2f){lo.w, hv.w};
        }
        // Prefetch next K-chunk while this one is consumed (global_prefetch).
        if (k0 + GM_KC < K) {
            __builtin_prefetch(A + (size_t)(m0 + (tid >> 2)) * K + k0 + GM_KC, 0, 3);
            __builtin_prefetch(B + (size_t)(k0 + GM_KC + (tid >> 4)) * N + n0 +
                                   ((tid & 15) << 3), 0, 3);
        }
        __syncthreads();

#pragma unroll
        for (int ks = 0; ks < GM_KC; ks += 4) {
            const int ka = ks + (hi << 1);       // even
            const int kp = (ks >> 1) + hi;       // pair index
            const v2f a = *(const v2f*)(&As[wm * 16 + l16][ka]);
#pragma unroll
            for (int nt = 0; nt < 4; ++nt) {
                const v2f b = Bs[kp][wn * 64 + nt * 16 + l16];
                acc[nt] = __builtin_amdgcn_wmma_f32_16x16x4_f32(
                    false, a, false, b, (short)0, acc[nt], false, false);
            }
        }
        __syncthreads();
    }

#pragma unroll
    for (int nt = 0; nt < 4; ++nt)
#pragma unroll
        for (int v = 0; v < 8; ++v) {
            const int m = m0 + wm * 16 + v + hi * 8;
            const int n = n0 + wn * 64 + nt * 16 + l16;
            C[(size_t)m * N + n] = acc[nt][v];
        }
}

// ---------------------------------------------------------------------------
// Flash attention forward (causal), fp32, HD=64.
// Block = 128 threads = 4 waves; block owns 64 query rows of one (b,h).
// qkv layout: [B, T, 3C] with q|k|v at col offsets 0|C|2C, head h at h*64.
// K tile stored row-major (S-phase B pairs are contiguous along d);
// V tile stored pair-interleaved (O-phase B pairs span key rows).
// ---------------------------------------------------------------------------
#define FA_LDW 68   // LDS row stride in floats (multiple of 4)

__global__ __launch_bounds__(128)
void attn_fwd_f32_wmma(const float* __restrict__ qkv, float* __restrict__ out,
                       int T, int Cdim) {
    __shared__ float Qs[64][FA_LDW];
    __shared__ float KVs[64][FA_LDW];   // K tile (row-major), reused for V (paired)
    __shared__ float Ps[64][FA_LDW];

    const int b    = blockIdx.z;
    const int h    = blockIdx.y;
    const int qb   = blockIdx.x;        // query block (64 rows)
    const int tid  = threadIdx.x;
    const int lane = tid & 31;
    const int wave = tid >> 5;          // 0..3, owns q rows [16*wave, +16)
    const int l16  = lane & 15;
    const int hi   = lane >> 4;
    const int C3   = 3 * Cdim;
    const size_t headBase = (size_t)b * T * C3 + (size_t)h * 64;

    v2f* const Vp = (v2f*)(&KVs[0][0]); // V as 32 row-pairs x FA_LDW v2f

    // Load Q tile 64x64 (8 float4 per thread).
    {
        const int r  = tid >> 1;
        const int cb = (tid & 1) * 32;
        const float* gq = qkv + headBase + (size_t)(qb * 64 + r) * C3;
#pragma unroll
        for (int i = 0; i < 8; ++i)
            *(float4*)(&Qs[r][cb + 4 * i]) = *(const float4*)(gq + cb + 4 * i);
    }

    v8f o[4];
#pragma unroll
    for (int i = 0; i < 4; ++i) o[i] = (v8f){};
    float mrow[8], lrow[8];
#pragma unroll
    for (int v = 0; v < 8; ++v) { mrow[v] = -1e30f; lrow[v] = 0.0f; }

    __syncthreads();

    const float scale = 0.125f;         // 1/sqrt(64)

    for (int j = 0; j <= qb; ++j) {
        // ---- Load K tile (64 keys x 64 dims), row-major ----
        {
            const int r  = tid >> 1;
            const int cb = (tid & 1) * 32;
            const float* gk = qkv + headBase + Cdim + (size_t)(j * 64 + r) * C3;
#pragma unroll
            for (int i = 0; i < 8; ++i)
                *(float4*)(&KVs[r][cb + 4 * i]) = *(const float4*)(gk + cb + 4 * i);
        }
        __syncthreads();

        // ---- S = Q * K^T : wave computes 16(q) x 64(k) ----
        v8f s[4];
#pragma unroll
        for (int i = 0; i < 4; ++i) s[i] = (v8f){};
#pragma unroll
        for (int ks = 0; ks < 64; ks += 4) {
            const int ka = ks + (hi << 1);
            const v2f a = *(const v2f*)(&Qs[wave * 16 + l16][ka]);
#pragma unroll
            for (int nt = 0; nt < 4; ++nt) {
                // B[d][key] = K[key][d]; pair contiguous along d.
                const v2f bb = *(const v2f*)(&KVs[nt * 16 + l16][ka]);
                s[nt] = __builtin_amdgcn_wmma_f32_16x16x4_f32(
                    false, a, false, bb, (short)0, s[nt], false, false);
            }
        }

        // ---- scale + causal mask ----
#pragma unroll
        for (int nt = 0; nt < 4; ++nt) {
            const int kg = j * 64 + nt * 16 + l16;
#pragma unroll
            for (int v = 0; v < 8; ++v) {
                const int qg = qb * 64 + wave * 16 + v + hi * 8;
                float x = s[nt][v] * scale;
                if (kg > qg) x = -1e30f;
                s[nt][v] = x;
            }
        }

        // ---- online softmax update (rows live across 16-lane halves) ----
#pragma unroll
        for (int v = 0; v < 8; ++v) {
            float mx = fmaxf(fmaxf(s[0][v], s[1][v]), fmaxf(s[2][v], s[3][v]));
#pragma unroll
            for (int off = 1; off < 16; off <<= 1)
                mx = fmaxf(mx, __shfl_xor(mx, off, 16));
            const float mnew  = fmaxf(mrow[v], mx);
            const float alpha = __expf(mrow[v] - mnew);
            mrow[v] = mnew;
            float rs = 0.0f;
#pragma unroll
            for (int nt = 0; nt < 4; ++nt) {
                const float p = __expf(s[nt][v] - mnew);
                s[nt][v] = p;
                rs += p;
            }
#pragma unroll
            for (int off = 1; off < 16; off <<= 1)
                rs += __shfl_xor(rs, off, 16);
            lrow[v] = lrow[v] * alpha + rs;
#pragma unroll
            for (int dt = 0; dt < 4; ++dt) o[dt][v] *= alpha;
        }

        // ---- spill P to LDS (C/D layout -> row-major [q][key]) ----
#pragma unroll
        for (int nt = 0; nt < 4; ++nt)
#pragma unroll
            for (int v = 0; v < 8; ++v)
                Ps[wave * 16 + v + hi * 8][nt * 16 + l16] = s[nt][v];
        __syncthreads();

        // ---- Load V tile pair-interleaved into KVs storage ----
        {
            const int p  = tid >> 2;            // 0..31 (key row pair)
            const int cb = (tid & 3) * 16;      // 0,16,32,48
            const float* g0 = qkv + headBase + 2 * Cdim +
                              (size_t)(j * 64 + 2 * p) * C3;
#pragma unroll
            for (int i = 0; i < 4; ++i) {
                const float4 lo = *(const float4*)(g0 + cb + 4 * i);
                const float4 hv = *(const float4*)(g0 + C3 + cb + 4 * i);
                Vp[p * FA_LDW + cb + 4 * i + 0] = (v2f){lo.x, hv.x};
                Vp[p * FA_LDW + cb + 4 * i + 1] = (v2f){lo.y, hv.y};
                Vp[p * FA_LDW + cb + 4 * i + 2] = (v2f){lo.z, hv.z};
                Vp[p * FA_LDW + cb + 4 * i + 3] = (v2f){lo.w, hv.w};
            }
        }
        __syncthreads();

        // ---- O += P * V : A = P[16 x 64], B = V[64 x 64] ----
#pragma unroll
        for (int ks = 0; ks < 64; ks += 4) {
            const int ka = ks + (hi << 1);
            const int kp = (ks >> 1) + hi;
            const v2f a = *(const v2f*)(&Ps[wave * 16 + l16][ka]);
#pragma unroll
            for (int dt = 0; dt < 4; ++dt) {
                const v2f bb = Vp[kp * FA_LDW + dt * 16 + l16];
                o[dt] = __builtin_amdgcn_wmma_f32_16x16x4_f32(
                    false, a, false, bb, (short)0, o[dt], false, false);
            }
        }
        __syncthreads();   // protect KVs/Ps before next iteration
    }

    // ---- normalize and write [B,T,C] ----
#pragma unroll
    for (int v = 0; v < 8; ++v) {
        const float inv = 1.0f / lrow[v];
        const int t = qb * 64 + wave * 16 + v + hi * 8;
#pragma unroll
        for (int dt = 0; dt < 4; ++dt) {
            out[((size_t)(b * T + t)) * Cdim + h * 64 + dt * 16 + l16] =
                o[dt][v] * inv;
        }
    }
}

// ---------------------------------------------------------------------------
// Launch: qkv GEMM -> flash attention -> output projection
// ---------------------------------------------------------------------------
extern "C" void kernel_launch(void* const* d_in, const int* in_sizes, int n_in,
                              void* d_out, int out_size, void* d_ws, size_t ws_size,
                              hipStream_t stream) {
    const float* x     = (const float*)d_in[0];   // [B,T,C]
    const float* W_qkv = (const float*)d_in[1];   // [C,3C]
    const float* W_out = (const float*)d_in[2];   // [C,C]
    float* out = (float*)d_out;                   // [B,T,C]

    const int Bsz = 4, T = 2048, C = 1024, H = 16;
    const int M = Bsz * T;                        // 8192

    float* qkv  = (float*)d_ws;                   // [M, 3C] = 96 MB
    float* attn = qkv + (size_t)M * 3 * C;        // [M, C]  = 32 MB

    // 1) qkv = x * W_qkv     (8192 x 3072 x 1024)
    gemm_f32_wmma<<<dim3((3 * C) / GM_TILE_N, M / GM_TILE_M), 256, 0, stream>>>(
        x, W_qkv, qkv, M, 3 * C, C);

    // 2) flash attention per (b, h), 64-query blocks
    attn_fwd_f32_wmma<<<dim3(T / 64, H, Bsz), 128, 0, stream>>>(qkv, attn, T, C);

    // 3) out = attn * W_out  (8192 x 1024 x 1024)
    gemm_f32_wmma<<<dim3(C / GM_TILE_N, M / GM_TILE_M), 256, 0, stream>>>(
        attn, W_out, out, M, C, C);
}